// BaselineGCN_55662776156293
// MI455X (gfx1250) — compile-verified
//
#include <hip/hip_runtime.h>
#include <hip/hip_bf16.h>

typedef float v2f __attribute__((ext_vector_type(2)));
typedef float v8f __attribute__((ext_vector_type(8)));

// ---------------------------------------------------------------------------
// deg[i] = 1 (self loop)
__global__ void gcn_init_deg(float* __restrict__ deg, int n) {
    int i = blockIdx.x * blockDim.x + threadIdx.x;
    if (i < n) deg[i] = 1.0f;
}

// deg[dst[e]] += 1
__global__ void gcn_deg_scatter(const int* __restrict__ dst, float* __restrict__ deg, int nE) {
    int e = blockIdx.x * blockDim.x + threadIdx.x;
    if (e < nE) atomicAdd(&deg[dst[e]], 1.0f);
}

// dis[i] = rsqrt(deg[i])  (deg >= 1 always, no zero-guard needed)
__global__ void gcn_rsqrt(float* __restrict__ deg, int n) {
    int i = blockIdx.x * blockDim.x + threadIdx.x;
    if (i < n) deg[i] = __frsqrt_rn(deg[i]);
}

// ---------------------------------------------------------------------------
// hs = (X @ W) * dis[row], duplicated into agg (self-loop init for scatter).
// X:[M,K] row-major, W:[K,N] row-major.
// One wave32 computes a 16-row x (NT*16)-col strip using
// V_WMMA_F32_16X16X4_F32, stepping K by 4 and reusing the A fragment across
// NT independent accumulators. All guards are branch-free (clamped address +
// cndmask zero) so the K-loop is pure load+wmma with EXEC all-1s throughout.
template <int NT>
__global__ void gcn_gemm_wmma(const float* __restrict__ X,
                              const float* __restrict__ W,
                              const float* __restrict__ dis,
                              float* __restrict__ hs,
                              float* __restrict__ agg,
                              int M, int K, int N) {
    int wave = (int)((blockIdx.x * blockDim.x + threadIdx.x) >> 5);
    int lane = threadIdx.x & 31;
    int m0 = wave << 4;
    if (m0 >= M) return;                 // wave-uniform exit

    int lo = lane & 15;                  // M index (A) / N index (B,D)
    int hi = lane >> 4;                  // K-pair select (A,B) / row-half (D)

    const float* xrow = X + (size_t)(m0 + lo) * K;

    int  col[NT];
    int  ccol[NT];
    bool ok[NT];
#pragma unroll
    for (int t = 0; t < NT; ++t) {
        col[t]  = t * 16 + lo;
        ok[t]   = col[t] < N;
        ccol[t] = ok[t] ? col[t] : 0;    // clamped: always a legal address
    }

    v8f c[NT];
#pragma unroll
    for (int t = 0; t < NT; ++t) c[t] = (v8f){};

    for (int k0 = 0; k0 < K; k0 += 4) {
        int ka = k0 + 2 * hi;
        // A 16x4: VGPR0 = K=2*hi, VGPR1 = K=2*hi+1, lane%16 = M  -> b64 load
        v2f a;
        a.x = xrow[ka];
        a.y = xrow[ka + 1];
        const float* w0 = W + (size_t)ka * N;
#pragma unroll
        for (int t = 0; t < NT; ++t) {
            // B 4x16: VGPR0 = K=2*hi, VGPR1 = K=2*hi+1, lane%16 = N
            float bx = w0[ccol[t]];          // unconditional loads,
            float by = w0[N + ccol[t]];      // value zeroed by cndmask
            v2f b;
            b.x = ok[t] ? bx : 0.0f;
            b.y = ok[t] ? by : 0.0f;
            c[t] = __builtin_amdgcn_wmma_f32_16x16x4_f32(
                       /*neg_a=*/false, a, /*neg_b=*/false, b,
                       /*c_mod=*/(short)0, c[t],
                       /*reuse_a=*/false, /*reuse_b=*/false);
        }
    }

    // D: VGPR r -> row m0 + 8*hi + r, col n0 + lane%16
    int rowBase = m0 + (hi << 3);
#pragma unroll
    for (int r = 0; r < 8; ++r) {
        int rr = rowBase + r;
        int rc = rr < M ? rr : M - 1;        // clamped load address
        float dr = dis[rc];
        size_t base = (size_t)rc * N;
#pragma unroll
        for (int t = 0; t < NT; ++t) {
            if (rr < M && ok[t]) {
                float v = c[t][r] * dr;
                hs[base + col[t]]  = v;
                agg[base + col[t]] = v;
            }
        }
    }
}

// ---------------------------------------------------------------------------
// agg[dst[e]] += hs[src[e]], float4 chunk per thread. L2-resident RMW.
__global__ void gcn_edge_scatter(const int* __restrict__ src,
                                 const int* __restrict__ dst,
                                 const float* __restrict__ hs,
                                 float* __restrict__ agg,
                                 int nE, int F, int chunks) {
    long long t = (long long)blockIdx.x * blockDim.x + threadIdx.x;
    long long total = (long long)nE * chunks;
    if (t >= total) return;
    int e = (int)(t / chunks);
    int c = (int)(t - (long long)e * chunks);
    int s = src[e];
    int d = dst[e];
    const float4 v = *(const float4*)(hs + (long long)s * F + (c << 2));
    float* p = agg + (long long)d * F + (c << 2);
    atomicAdd(p + 0, v.x);
    atomicAdd(p + 1, v.y);
    atomicAdd(p + 2, v.z);
    atomicAdd(p + 3, v.w);
}

// ---------------------------------------------------------------------------
// out[i][f] = (relu?)(dis[i] * agg[i][f] + bias[f]); supports out == agg.
__global__ void gcn_finalize(const float* __restrict__ agg,
                             const float* __restrict__ dis,
                             const float* __restrict__ bias,
                             float* __restrict__ out,
                             int nNodes, int F, int doRelu) {
    long long t = (long long)blockIdx.x * blockDim.x + threadIdx.x;
    if (t >= (long long)nNodes * F) return;
    int i = (int)(t / F);
    int f = (int)(t - (long long)i * F);
    float v = fmaf(dis[i], agg[t], bias[f]);
    if (doRelu) v = fmaxf(v, 0.0f);
    out[t] = v;
}

// ---------------------------------------------------------------------------
static inline void launch_gemm(int tilesN, const float* X, const float* W,
                               const float* dis, float* hs, float* agg,
                               int M, int K, int N, hipStream_t stream) {
    int mt = (M + 15) / 16;
    long long thr = (long long)mt * 32;
    int blocks = (int)((thr + 255) / 256);
    switch (tilesN) {
    case 1: gcn_gemm_wmma<1><<<blocks, 256, 0, stream>>>(X, W, dis, hs, agg, M, K, N); break;
    case 2: gcn_gemm_wmma<2><<<blocks, 256, 0, stream>>>(X, W, dis, hs, agg, M, K, N); break;
    case 3: gcn_gemm_wmma<3><<<blocks, 256, 0, stream>>>(X, W, dis, hs, agg, M, K, N); break;
    default: gcn_gemm_wmma<4><<<blocks, 256, 0, stream>>>(X, W, dis, hs, agg, M, K, N); break;
    }
}

extern "C" void kernel_launch(void* const* d_in, const int* in_sizes, int n_in,
                              void* d_out, int out_size, void* d_ws, size_t ws_size,
                              hipStream_t stream) {
    const float* x  = (const float*)d_in[0];   // [N, 128]
    const int*   ei = (const int*)d_in[1];     // [2, E] (row 0 = src, row 1 = dst)
    const float* W1 = (const float*)d_in[2];   // [128, 64]
    const float* b1 = (const float*)d_in[3];   // [64]
    const float* W2 = (const float*)d_in[4];   // [64, 40]
    const float* b2 = (const float*)d_in[5];   // [40]
    float* out = (float*)d_out;                // [N, 40]

    const int IN  = 128;
    const int HID = in_sizes[3];               // 64
    const int OUT = in_sizes[5];               // 40
    const int N   = in_sizes[0] / IN;          // 100000
    const int E   = in_sizes[1] / 2;           // 3200000

    const int* src = ei;
    const int* dst = ei + E;

    // Workspace layout (floats), 256B-aligned blocks
    float* dis  = (float*)d_ws;                        // N (reused: deg -> dis)
    float* hs1  = dis + (((size_t)N + 63) & ~63ull);   // N*HID
    float* agg1 = hs1 + (size_t)N * HID;               // N*HID
    float* hs2  = agg1 + (size_t)N * HID;              // N*OUT
    // layer-2 agg buffer is d_out itself (finalize2 is in-place)

    const int B = 256;

    // --- normalization coefficients ------------------------------------
    gcn_init_deg<<<(N + B - 1) / B, B, 0, stream>>>(dis, N);
    gcn_deg_scatter<<<(E + B - 1) / B, B, 0, stream>>>(dst, dis, E);
    gcn_rsqrt<<<(N + B - 1) / B, B, 0, stream>>>(dis, N);

    // --- layer 1 --------------------------------------------------------
    launch_gemm((HID + 15) / 16, x, W1, dis, hs1, agg1, N, IN, HID, stream);

    int chunks1 = HID / 4;                              // 16
    long long sthr1 = (long long)E * chunks1;
    gcn_edge_scatter<<<(int)((sthr1 + B - 1) / B), B, 0, stream>>>(
        src, dst, hs1, agg1, E, HID, chunks1);

    long long fthr1 = (long long)N * HID;
    // relu(dis*agg1 + b1) -> reuse hs1 as the layer-1 activation h1
    gcn_finalize<<<(int)((fthr1 + B - 1) / B), B, 0, stream>>>(
        agg1, dis, b1, hs1, N, HID, /*relu=*/1);

    // --- layer 2 --------------------------------------------------------
    launch_gemm((OUT + 15) / 16, hs1, W2, dis, hs2, out, N, HID, OUT, stream);

    int chunks2 = OUT / 4;                              // 10
    long long sthr2 = (long long)E * chunks2;
    gcn_edge_scatter<<<(int)((sthr2 + B - 1) / B), B, 0, stream>>>(
        src, dst, hs2, out, E, OUT, chunks2);

    long long fthr2 = (long long)N * OUT;
    gcn_finalize<<<(int)((fthr2 + B - 1) / B), B, 0, stream>>>(
        out, dis, b2, out, N, OUT, /*relu=*/0);
}